// Q_Model_43782896615432
// MI455X (gfx1250) — compile-verified
//
#include <hip/hip_runtime.h>
#include <math.h>

typedef __attribute__((ext_vector_type(2))) float v2f;
typedef __attribute__((ext_vector_type(8))) float v8f;

// Branchless float cos: Cody-Waite pi/2 reduction + minimax polys on [-pi/4, pi/4].
// Accurate to ~1 ulp for the |x| <= ~1e3 range this kernel produces; no slow path.
__device__ __forceinline__ float fast_cosf(float x) {
    const float xn = rintf(x * 0.63661977f);          // round(x * 2/pi)
    const int   k  = (int)xn;
    float r = fmaf(xn, -1.57079637e+0f, x);           // x - n*fl(pi/2)
    r       = fmaf(xn,  4.37113883e-8f, r);           // + n*(fl(pi/2) - pi/2)
    const float s = r * r;
    // cos poly on [-pi/4, pi/4]
    float c = 2.44331571e-5f;
    c = fmaf(c, s, -1.38873163e-3f);
    c = fmaf(c, s,  4.16666456e-2f);
    c = fmaf(c, s, -5.00000000e-1f);
    c = fmaf(c, s,  1.00000000e+0f);
    // sin poly on [-pi/4, pi/4]
    float p = -1.95152959e-4f;
    p = fmaf(p, s,  8.33216087e-3f);
    p = fmaf(p, s, -1.66666546e-1f);
    const float sn = fmaf(p * s, r, r);
    // quadrant select: cos, -sin, -cos, sin for k mod 4 = 0,1,2,3
    float v = (k & 1) ? sn : c;
    return ((k + 1) & 2) ? -v : v;
}

// One wave32 processes 16 samples:
//   theta(16x6) = [sen|1]·Wsen^T + [sa|1]·Wsar^T   via 2x v_wmma_f32_16x16x4_f32 (exact fp32)
//   c = cos(theta); closed-form circuit products; tiny output affine in VALU.
__global__ __launch_bounds__(256) void qexp_kernel(
    const float* __restrict__ sa,
    const float* __restrict__ sen,
    const float* __restrict__ W_sar,   const float* __restrict__ b_sar,
    const float* __restrict__ W_sen,   const float* __restrict__ b_sen,
    const float* __restrict__ W_resar, const float* __restrict__ b_resar,
    const float* __restrict__ W_resen, const float* __restrict__ b_resen,
    float* __restrict__ out, const int Bn)
{
    const int lane = threadIdx.x & 31;
    const int wid  = blockIdx.x * (blockDim.x >> 5) + (threadIdx.x >> 5);
    const long base = (long)wid << 4;           // first sample of this wave's 16-row tile
    if (base >= (long)Bn) return;               // wave-uniform guard (EXEC stays full)

    const int n  = lane & 15;                   // A: row m ; B/D: column n
    const int kh = lane >> 4;                   // K half: 0 -> K=0,1 ; 1 -> K=2,3

    long s = base + n;
    if (s >= (long)Bn) s = (long)Bn - 1;        // clamp reads for ragged tail

    // ---- A tiles (16x4 fp32): row = [f0, f1, f2, 1]
    v2f a1, a2;
    if (kh == 0) {
        a1.x = sen[3*s+0]; a1.y = sen[3*s+1];
        a2.x = sa [3*s+0]; a2.y = sa [3*s+1];
    } else {
        a1.x = sen[3*s+2]; a1.y = 1.0f;
        a2.x = sa [3*s+2]; a2.y = 1.0f;
    }

    // ---- B tiles (4x16 fp32): lane = column n, VGPR pair = K = {2*kh, 2*kh+1}
    // B1 cols 0..2 = W_sen rows + bias at K=3; B2 cols 3..5 = W_sar rows + bias; rest zero.
    v2f b1 = {0.0f, 0.0f};
    v2f b2 = {0.0f, 0.0f};
    if (n < 3) {
        if (kh == 0) { b1.x = W_sen[3*n+0]; b1.y = W_sen[3*n+1]; }
        else         { b1.x = W_sen[3*n+2]; b1.y = b_sen[n];     }
    } else if (n < 6) {
        const int j = n - 3;
        if (kh == 0) { b2.x = W_sar[3*j+0]; b2.y = W_sar[3*j+1]; }
        else         { b2.x = W_sar[3*j+2]; b2.y = b_sar[j];     }
    }

    // theta tile: D[m][n], columns 0..5 valid (others exactly 0 -> cos = 1, harmless)
    v8f th = {};
    th = __builtin_amdgcn_wmma_f32_16x16x4_f32(false, a1, false, b1, (short)0, th, false, false);
    th = __builtin_amdgcn_wmma_f32_16x16x4_f32(false, a2, false, b2, (short)0, th, false, false);

    // Per-lane output-layer coefficients, selected by this lane's D column:
    //   cols 0,1 -> sar rows (use x3,x4,x5); cols 2,3,4 -> sen rows (use x0,x1,x2)
    float w0 = 0.f, w1 = 0.f, w2 = 0.f, bb = 0.f;
    const bool isSar = (n < 2);
    if (n < 2) {
        w0 = W_resar[3*n+0]; w1 = W_resar[3*n+1]; w2 = W_resar[3*n+2]; bb = b_resar[n];
    } else if (n < 5) {
        const int j = n - 2;
        w0 = W_resen[3*j+0]; w1 = W_resen[3*j+1]; w2 = W_resen[3*j+2]; bb = b_resen[j];
    }

    const int  hb     = lane & 16;              // base lane of this half-wave
    const long senOff = 2L * (long)Bn;          // d_out = [sar (B,2) | sen_out (B,3)] flat

    #pragma unroll
    for (int r = 0; r < 8; ++r) {
        // D layout: this VGPR holds row m = r + 8*kh for column n
        const float c  = fast_cosf(th[r]);
        // broadcast the 6 per-wire cosines of this row set across the half-wave
        const float c0 = __shfl(c, hb + 0);
        const float c1 = __shfl(c, hb + 1);
        const float c2 = __shfl(c, hb + 2);
        const float c3 = __shfl(c, hb + 3);
        const float c4 = __shfl(c, hb + 4);
        const float c5 = __shfl(c, hb + 5);
        // closed-form circuit expectations
        const float p01 = c0*c1, p12 = c1*c2, p012 = p01*c2;   // x1, x0, x2
        const float p34 = c3*c4, p45 = c4*c5, p345 = p34*c5;   // x4, x3, x5
        const float xA = isSar ? p45  : p12;
        const float xB = isSar ? p34  : p01;
        const float xC = isSar ? p345 : p012;
        const float o  = fmaf(w0, xA, fmaf(w1, xB, fmaf(w2, xC, bb)));

        const long sm = base + r + 8*kh;
        if (sm < (long)Bn) {
            if (n < 2)      out[2*sm + n]              = o;   // sar
            else if (n < 5) out[senOff + 3*sm + (n-2)] = o;   // sen_out
        }
    }
}

extern "C" void kernel_launch(void* const* d_in, const int* in_sizes, int n_in,
                              void* d_out, int out_size, void* d_ws, size_t ws_size,
                              hipStream_t stream) {
    const float* sa      = (const float*)d_in[0];
    const float* sen     = (const float*)d_in[1];
    const float* W_sar   = (const float*)d_in[2];
    const float* b_sar   = (const float*)d_in[3];
    const float* W_sen   = (const float*)d_in[4];
    const float* b_sen   = (const float*)d_in[5];
    const float* W_resar = (const float*)d_in[6];
    const float* b_resar = (const float*)d_in[7];
    const float* W_resen = (const float*)d_in[8];
    const float* b_resen = (const float*)d_in[9];
    float* out = (float*)d_out;

    const int Bn     = in_sizes[0] / 3;         // sa is (B,3)
    const int waves  = (Bn + 15) / 16;          // 16 samples per wave
    const int blocks = (waves + 7) / 8;         // 8 waves (256 threads) per block
    qexp_kernel<<<blocks, 256, 0, stream>>>(sa, sen, W_sar, b_sar, W_sen, b_sen,
                                            W_resar, b_resar, W_resen, b_resen, out, Bn);
}